// DenseGCN_21045339750898
// MI455X (gfx1250) — compile-verified
//
#include <hip/hip_runtime.h>
#include <hip/hip_bf16.h>

typedef __attribute__((ext_vector_type(2))) float v2f;
typedef __attribute__((ext_vector_type(8))) float v8f;

#define N_NODES 50000
#define N_EDGES 800000
#define FIN 256
#define FH 96
#define FC 40
#define FC_PAD 48   // 3 full 16-wide WMMA tiles

// ---------------------------------------------------------------- utilities
__global__ void fill_kernel(float* __restrict__ p, float v, long n) {
    long i = (long)blockIdx.x * blockDim.x + threadIdx.x;
    if (i < n) p[i] = v;
}

// Zero-pad W2 [40,96] -> W2p [48,96] (rows contiguous, so flat copy + zero tail)
__global__ void pad_w2_kernel(const float* __restrict__ W2, float* __restrict__ W2p,
                              int nsrc, int ntot) {
    int i = blockIdx.x * blockDim.x + threadIdx.x;
    if (i < ntot) W2p[i] = (i < nsrc) ? W2[i] : 0.0f;
}

// deg initialized to 1.0 (self loop); add 1 per occurrence in row[]
__global__ void degree_kernel(const long long* __restrict__ row,
                              float* __restrict__ deg, int E) {
    int e = blockIdx.x * blockDim.x + threadIdx.x;
    if (e < E) atomicAdd(&deg[(int)row[e]], 1.0f);
}

// deg >= 1 always (self loops), so dinv = deg^-0.5 matches the reference where()
__global__ void dinv_kernel(float* __restrict__ deg, int n) {
    int i = blockIdx.x * blockDim.x + threadIdx.x;
    if (i < n) deg[i] = rsqrtf(deg[i]);
}

// ------------------------------------------------ fp32 WMMA strip GEMM
// D = act(A [+ bias]) * W^T.  A:[M,K] row-major, W:[NT*16,K] row-major
// (zero-padded to a tile multiple), D:[M,NC] row-major.
// One wave32 owns a 16-row strip and all NT column tiles: each A-fragment
// load feeds NT V_WMMA_F32_16X16X4_F32 ops (NT*8 accumulator VGPRs).
// FUSE_RELU_BIAS fuses y = relu(A + bias[k]) into the A-fragment load.
template <int NT, int NC, bool FUSE_RELU_BIAS>
__global__ void gemm_wmma_strip(const float* __restrict__ A,
                                const float* __restrict__ bin,
                                const float* __restrict__ W,
                                float* __restrict__ D,
                                int M, int K) {
    const int lane = threadIdx.x & 31;
    const int wave = threadIdx.x >> 5;
    const int mt = blockIdx.x * (blockDim.x >> 5) + wave;
    if (mt >= (M >> 4)) return;           // wave-uniform: EXEC stays all-1s

    const int m0    = mt << 4;
    const int l15   = lane & 15;
    const int khalf = (lane >> 4) << 1;   // f32 A/B frag: lanes 0-15 K={0,1}, 16-31 K={2,3}

    const float* ap = A + (long)(m0 + l15) * K + khalf;
    const float* wp = W + (long)l15 * K + khalf;

    v8f c[NT];
    v8f zero = {};
    #pragma unroll
    for (int t = 0; t < NT; ++t) c[t] = zero;

    for (int kk = 0; kk < K; kk += 4) {
        float a0 = ap[kk];
        float a1 = ap[kk + 1];
        if constexpr (FUSE_RELU_BIAS) {
            a0 = fmaxf(a0 + bin[kk + khalf],     0.0f);
            a1 = fmaxf(a1 + bin[kk + khalf + 1], 0.0f);
        }
        v2f a; a.x = a0; a.y = a1;
        #pragma unroll
        for (int t = 0; t < NT; ++t) {
            v2f b;
            b.x = wp[(long)t * 16 * K + kk];
            b.y = wp[(long)t * 16 * K + kk + 1];
            // (neg_a, A, neg_b, B, c_mod, C, reuse_a, reuse_b)
            c[t] = __builtin_amdgcn_wmma_f32_16x16x4_f32(false, a, false, b,
                                                         (short)0, c[t], false, false);
        }
    }

    // C/D layout: VGPR g -> row m0+g (lanes 0-15) / m0+g+8 (lanes 16-31)
    const int mbase = m0 + ((lane >> 4) << 3);
    #pragma unroll
    for (int t = 0; t < NT; ++t) {
        const int n = t * 16 + l15;
        if (n < NC) {
            #pragma unroll
            for (int g = 0; g < 8; ++g)
                D[(long)(mbase + g) * NC + n] = c[t][g];
        }
    }
}

// ----------------------------------------- per-edge message + scatter-add (L2)
// blockDim.x = F lanes (one per feature), blockDim.y edges per block.
// Edges e >= E are the self-loops (r = c = e - E, weight 1.0).
__global__ void scatter_kernel(const long long* __restrict__ row,
                               const long long* __restrict__ col,
                               const float* __restrict__ ew,
                               const float* __restrict__ dinv,
                               const float* __restrict__ h,
                               float* __restrict__ agg,
                               int E, int Etot, int F) {
    int e = blockIdx.x * blockDim.y + threadIdx.y;
    if (e >= Etot) return;
    int f = threadIdx.x;
    int r, c; float w;
    if (e < E) { r = (int)row[e]; c = (int)col[e]; w = ew[e]; }
    else       { r = c = e - E;   w = 1.0f; }
    float nrm = dinv[r] * w * dinv[c];
    atomicAdd(&agg[(long)c * F + f], nrm * h[(long)r * F + f]);
}

// -------------------------------------- fused bias + log_softmax over 40 cols
// One wave32 per node: lane carries cols {lane, lane+32}.
__global__ void logsoftmax_kernel(const float* __restrict__ agg2,
                                  const float* __restrict__ b2,
                                  float* __restrict__ out, int N) {
    const int lane = threadIdx.x & 31;
    const int wave = threadIdx.x >> 5;
    const int node = blockIdx.x * (blockDim.x >> 5) + wave;
    if (node >= N) return;
    const float* rp = agg2 + (long)node * FC;
    float v1 = rp[lane] + b2[lane];
    float v2 = (lane < FC - 32) ? (rp[32 + lane] + b2[32 + lane]) : -INFINITY;

    float m = fmaxf(v1, v2);
    #pragma unroll
    for (int off = 16; off; off >>= 1) m = fmaxf(m, __shfl_xor(m, off, 32));

    float s = expf(v1 - m) + ((lane < FC - 32) ? expf(v2 - m) : 0.0f);
    #pragma unroll
    for (int off = 16; off; off >>= 1) s += __shfl_xor(s, off, 32);

    float lse = m + logf(s);
    out[(long)node * FC + lane] = v1 - lse;
    if (lane < FC - 32) out[(long)node * FC + 32 + lane] = v2 - lse;
}

// ---------------------------------------------------------------- launch
extern "C" void kernel_launch(void* const* d_in, const int* in_sizes, int n_in,
                              void* d_out, int out_size, void* d_ws, size_t ws_size,
                              hipStream_t stream) {
    const float*     x   = (const float*)d_in[0];
    const long long* ei  = (const long long*)d_in[1];   // int64 [2, E]
    const long long* row = ei;
    const long long* col = ei + N_EDGES;
    const float*     ew  = (const float*)d_in[2];
    const float*     W1  = (const float*)d_in[3];
    const float*     b1  = (const float*)d_in[4];
    const float*     W2  = (const float*)d_in[5];
    const float*     b2  = (const float*)d_in[6];
    float*           out = (float*)d_out;

    // workspace carve-up (all re-initialized every call; graph-replay safe)
    float* dinv = (float*)d_ws;                       // N
    float* h1   = dinv + N_NODES;                     // N*96
    float* agg1 = h1   + (size_t)N_NODES * FH;        // N*96
    float* h2   = agg1 + (size_t)N_NODES * FH;        // N*40
    float* agg2 = h2   + (size_t)N_NODES * FC;        // N*40
    float* w2p  = agg2 + (size_t)N_NODES * FC;        // 48*96 zero-padded W2

    const int Etot = N_EDGES + N_NODES;

    // init: deg starts at 1.0 (self loop), accumulators at 0, pad W2
    fill_kernel<<<(N_NODES + 255) / 256, 256, 0, stream>>>(dinv, 1.0f, N_NODES);
    fill_kernel<<<((long)N_NODES * FH + 255) / 256, 256, 0, stream>>>(agg1, 0.0f, (long)N_NODES * FH);
    fill_kernel<<<((long)N_NODES * FC + 255) / 256, 256, 0, stream>>>(agg2, 0.0f, (long)N_NODES * FC);
    pad_w2_kernel<<<(FC_PAD * FH + 255) / 256, 256, 0, stream>>>(W2, w2p, FC * FH, FC_PAD * FH);

    degree_kernel<<<(N_EDGES + 255) / 256, 256, 0, stream>>>(row, dinv, N_EDGES);
    dinv_kernel<<<(N_NODES + 255) / 256, 256, 0, stream>>>(dinv, N_NODES);

    // layer 1: h1 = x @ W1^T   (one wave per 16-row strip, 6 col-tiles/wave)
    {
        int strips = N_NODES / 16;                 // 3125
        gemm_wmma_strip<FH / 16, FH, false>
            <<<(strips + 7) / 8, 256, 0, stream>>>(x, nullptr, W1, h1, N_NODES, FIN);
    }
    {
        dim3 blk(FH, 4);                           // 384 threads, 96 lanes/edge
        scatter_kernel<<<(Etot + 3) / 4, blk, 0, stream>>>(row, col, ew, dinv,
                                                           h1, agg1, N_EDGES, Etot, FH);
    }

    // layer 2: h2 = relu(agg1 + b1) @ W2p^T   (bias+ReLU fused into A-frag load)
    {
        int strips = N_NODES / 16;
        gemm_wmma_strip<FC_PAD / 16, FC, true>
            <<<(strips + 7) / 8, 256, 0, stream>>>(agg1, b1, w2p, h2, N_NODES, FH);
    }
    {
        dim3 blk(FC, 8);                           // 320 threads, 40 lanes/edge
        scatter_kernel<<<(Etot + 7) / 8, blk, 0, stream>>>(row, col, ew, dinv,
                                                           h2, agg2, N_EDGES, Etot, FC);
    }

    // fused bias + log_softmax, one wave per node
    logsoftmax_kernel<<<(N_NODES + 7) / 8, 256, 0, stream>>>(agg2, b2, out, N_NODES);
}